// CrossAttention_39591008534830
// MI455X (gfx1250) — compile-verified
//
#include <hip/hip_runtime.h>
#include <math.h>

typedef __attribute__((ext_vector_type(2))) float v2f;
typedef __attribute__((ext_vector_type(8))) float v8f;

// D = A(16x4 f32) * B(4x16 f32) + C(16x16 f32), full fp32 precision on the matrix pipe.
#define WMMA_F32(a, b, c) \
  __builtin_amdgcn_wmma_f32_16x16x4_f32(false, (a), false, (b), (short)0, (c), false, false)

// ---------------------------------------------------------------------------
// Kernel 1: fused projections  q = query@Wq, k = key@Wk, v = key@Wv
// One wave -> one 16-row x 64-col tile of one projection. K=128 -> 32 slices.
// 3 matrices x 1024 row-tiles = 3072 waves = 384 blocks x 8 waves.
// ---------------------------------------------------------------------------
__global__ __launch_bounds__(256) void ca_proj_kernel(
    const float* __restrict__ query, const float* __restrict__ key,
    const float* __restrict__ Wq, const float* __restrict__ Wk,
    const float* __restrict__ Wv,
    float* __restrict__ qproj, float* __restrict__ kproj,
    float* __restrict__ vproj) {
  const int lane = threadIdx.x & 31;
  const int wave = threadIdx.x >> 5;
  const int gw   = blockIdx.x * 8 + wave;   // 0..3071
  const int mat  = gw >> 10;                // 0=q, 1=k, 2=v
  const int tile = gw & 1023;               // 0..1023 row tiles (B*2048/16)
  const int rowbase = tile << 4;
  const int l16 = lane & 15;
  const int h   = lane >> 4;                // lane half

  const float* X = (mat == 0) ? query : key;
  const float* W = (mat == 0) ? Wq : (mat == 1 ? Wk : Wv);
  float*       Y = (mat == 0) ? qproj : (mat == 1 ? kproj : vproj);

  v8f acc0 = {}, acc1 = {}, acc2 = {}, acc3 = {};

  // A-format: lane holds row (rowbase+l16), cols 4s+2h, 4s+2h+1 (8B aligned).
  const float* xrow = X + (size_t)(rowbase + l16) * 128 + 2 * h;
  for (int s = 0; s < 32; ++s) {
    v2f a = *(const v2f*)(xrow + 4 * s);
    // B-format: lane holds W[4s+2h + v][16*nt + l16], v = 0,1.
    const float* wr = W + (size_t)(4 * s + 2 * h) * 64 + l16;
    { v2f b; b.x = wr[0];  b.y = wr[64];      acc0 = WMMA_F32(a, b, acc0); }
    { v2f b; b.x = wr[16]; b.y = wr[64 + 16]; acc1 = WMMA_F32(a, b, acc1); }
    { v2f b; b.x = wr[32]; b.y = wr[64 + 32]; acc2 = WMMA_F32(a, b, acc2); }
    { v2f b; b.x = wr[48]; b.y = wr[64 + 48]; acc3 = WMMA_F32(a, b, acc3); }
  }

  // C-format store: row r+8h, col 16*nt + l16.
  float* yb = Y + (size_t)(rowbase + 8 * h) * 64 + l16;
#pragma unroll
  for (int r = 0; r < 8; ++r) {
    yb[(size_t)r * 64 + 0]  = acc0[r];
    yb[(size_t)r * 64 + 16] = acc1[r];
    yb[(size_t)r * 64 + 32] = acc2[r];
    yb[(size_t)r * 64 + 48] = acc3[r];
  }
}

// ---------------------------------------------------------------------------
// Kernel 2: flash attention. One wave owns 16 query rows, online softmax over
// 128 key-blocks of 16. 1024 waves = 256 blocks x 4 waves. Per-wave 1KB LDS
// patch used for the C-layout -> A-layout transpose of P.
// ---------------------------------------------------------------------------
__global__ __launch_bounds__(128) void ca_flash_kernel(
    const float* __restrict__ qproj, const float* __restrict__ kproj,
    const float* __restrict__ vproj, float* __restrict__ out) {
  __shared__ float lds_p[4 * 256];

  const int lane = threadIdx.x & 31;
  const int wave = threadIdx.x >> 5;
  const int gw    = blockIdx.x * 4 + wave;  // 0..1023
  const int b     = gw >> 7;                // batch
  const int qbase = (gw & 127) << 4;        // query row base
  const int l16 = lane & 15;
  const int h   = lane >> 4;

  const float* Qb = qproj + (size_t)b * 2048 * 64;
  const float* Kb = kproj + (size_t)b * 2048 * 64;
  const float* Vb = vproj + (size_t)b * 2048 * 64;

  // Q tile in A-format, kept resident: qa[s] = Q[qbase+l16][4s+2h .. +1]
  v2f qa[16];
  {
    const float* qrow = Qb + (size_t)(qbase + l16) * 64 + 2 * h;
#pragma unroll
    for (int s = 0; s < 16; ++s) qa[s] = *(const v2f*)(qrow + 4 * s);
  }

  float mrow[8], lrow[8];
#pragma unroll
  for (int r = 0; r < 8; ++r) { mrow[r] = -INFINITY; lrow[r] = 0.0f; }
  v8f o0 = {}, o1 = {}, o2 = {}, o3 = {};

  float* myP = lds_p + wave * 256;

  for (int kb = 0; kb < 2048; kb += 16) {
    if (kb + 16 < 2048) {  // prefetch next K/V block (global_prefetch_b8)
      __builtin_prefetch(Kb + (size_t)(kb + 16 + l16) * 64, 0, 0);
      __builtin_prefetch(Vb + (size_t)(kb + 16 + l16) * 64, 0, 0);
    }

    // ---- S = Q Kt : B-format K load has the same per-lane pattern as A.
    const float* krow = Kb + (size_t)(kb + l16) * 64 + 2 * h;
    v8f c = {};
#pragma unroll
    for (int s = 0; s < 16; ++s) {
      v2f kv = *(const v2f*)(krow + 4 * s);
      c = WMMA_F32(qa[s], kv, c);
    }

    // ---- online softmax in C-layout: row r+8h lives in this lane-half.
    float cm[8];
#pragma unroll
    for (int r = 0; r < 8; ++r) cm[r] = c[r];
#pragma unroll
    for (int m = 1; m <= 8; m <<= 1)
#pragma unroll
      for (int r = 0; r < 8; ++r) cm[r] = fmaxf(cm[r], __shfl_xor(cm[r], m, 32));

    float alpha[8], p[8], ps[8];
#pragma unroll
    for (int r = 0; r < 8; ++r) {
      float mn = fmaxf(mrow[r], cm[r]);
      alpha[r] = __expf(mrow[r] - mn);   // 0 on first block (exp(-inf))
      mrow[r]  = mn;
      p[r]  = __expf(c[r] - mn);
      ps[r] = p[r];
    }
#pragma unroll
    for (int m = 1; m <= 8; m <<= 1)
#pragma unroll
      for (int r = 0; r < 8; ++r) ps[r] += __shfl_xor(ps[r], m, 32);
#pragma unroll
    for (int r = 0; r < 8; ++r) lrow[r] = lrow[r] * alpha[r] + ps[r];

    // rescale accumulators
#pragma unroll
    for (int r = 0; r < 8; ++r) {
      o0[r] *= alpha[r]; o1[r] *= alpha[r]; o2[r] *= alpha[r]; o3[r] *= alpha[r];
    }

    // ---- transpose P (C-layout -> A-layout) through private LDS patch.
    // Same-wave LDS is in-order; s_wait_dscnt makes cross-lane stores visible.
#pragma unroll
    for (int r = 0; r < 8; ++r) myP[(r + 8 * h) * 16 + l16] = p[r];
    asm volatile("s_wait_dscnt 0" ::: "memory");
    v2f pa[4];
#pragma unroll
    for (int s = 0; s < 4; ++s)
      pa[s] = *(const v2f*)(myP + l16 * 16 + 4 * s + 2 * h);

    // ---- O += P V : B-format V, K = key offset 4s+2h+v, N = 16*nt + l16.
    const float* vr = Vb + (size_t)kb * 64 + l16;
#pragma unroll
    for (int s = 0; s < 4; ++s) {
      const float* v0 = vr + (size_t)(4 * s + 2 * h) * 64;
      { v2f bv; bv.x = v0[0];  bv.y = v0[64];      o0 = WMMA_F32(pa[s], bv, o0); }
      { v2f bv; bv.x = v0[16]; bv.y = v0[64 + 16]; o1 = WMMA_F32(pa[s], bv, o1); }
      { v2f bv; bv.x = v0[32]; bv.y = v0[64 + 32]; o2 = WMMA_F32(pa[s], bv, o2); }
      { v2f bv; bv.x = v0[48]; bv.y = v0[64 + 48]; o3 = WMMA_F32(pa[s], bv, o3); }
    }
  }

  // ---- epilogue: out = O / l, C-layout store.
  float* ob = out + ((size_t)b * 2048 + qbase + 8 * h) * 64 + l16;
#pragma unroll
  for (int r = 0; r < 8; ++r) {
    float inv = 1.0f / lrow[r];
    ob[(size_t)r * 64 + 0]  = o0[r] * inv;
    ob[(size_t)r * 64 + 16] = o1[r] * inv;
    ob[(size_t)r * 64 + 32] = o2[r] * inv;
    ob[(size_t)r * 64 + 48] = o3[r] * inv;
  }
}

// ---------------------------------------------------------------------------
extern "C" void kernel_launch(void* const* d_in, const int* in_sizes, int n_in,
                              void* d_out, int out_size, void* d_ws,
                              size_t ws_size, hipStream_t stream) {
  const float* query = (const float*)d_in[0];  // [8, 2048, 128]
  const float* key   = (const float*)d_in[1];  // [8, 2048, 128]
  const float* Wq    = (const float*)d_in[2];  // [128, 64]
  const float* Wk    = (const float*)d_in[3];  // [128, 64]
  const float* Wv    = (const float*)d_in[4];  // [128, 64]
  float* out = (float*)d_out;                  // [8, 2048, 64]

  const size_t projElems = (size_t)8 * 2048 * 64;  // 1M floats = 4MB each
  float* qproj = (float*)d_ws;
  float* kproj = qproj + projElems;
  float* vproj = kproj + projElems;

  ca_proj_kernel<<<384, 256, 0, stream>>>(query, key, Wq, Wk, Wv,
                                          qproj, kproj, vproj);
  ca_flash_kernel<<<256, 128, 0, stream>>>(qproj, kproj, vproj, out);
}